// RTM3DLoss_12421045420828
// MI455X (gfx1250) — compile-verified
//
#include <hip/hip_runtime.h>

// RTM3D / CenterNet Gaussian heatmap rendering via scatter-max.
// Strategy:
//  - hm_zero: vectorized zero-fill of the 47 MB output (fits in 192 MB L2).
//  - hm_scatter: 1 block (16 wave32) per instance. The 2D Gaussian patch is
//    separable, so each 16x16 tile is a rank-1 outer product computed with a
//    single V_WMMA_F32_16X16X4_F32 (A = 1-D gaussian column in K=0, B = 1-D
//    gaussian row in K=0, other K slots zero). Scatter-max via u32 atomicMax
//    (bit pattern of non-negative f32 is order-preserving).

typedef __attribute__((ext_vector_type(2))) float v2f;
typedef __attribute__((ext_vector_type(8))) float v8f;

#define HM_B 32
#define HM_C 3
#define HM_H 192
#define HM_W 640
#define MAXR 30

__global__ __launch_bounds__(256) void hm_zero(float4* __restrict__ out, int n4) {
  int i = blockIdx.x * blockDim.x + threadIdx.x;
  if (i < n4) out[i] = make_float4(0.f, 0.f, 0.f, 0.f);
}

__device__ __forceinline__ float gauss1d(int t, float radius, float neg_inv2s2) {
  float ft = (float)t;
  float g = __expf(ft * ft * neg_inv2s2);
  // separable half of the (|dx|<=r & |dy|<=r) box mask
  return (fabsf(ft) <= radius) ? g : 0.0f;
}

__global__ __launch_bounds__(512) void hm_scatter(
    const float* __restrict__ bboxes, const int* __restrict__ classes,
    const int* __restrict__ img_id, const unsigned char* __restrict__ noise_mask,
    float* __restrict__ out) {
  const int i    = blockIdx.x;            // instance
  const int lane = threadIdx.x & 31;
  const int wid  = threadIdx.x >> 5;      // 16 waves -> 4x4 tile grid
  const int dy0  = -MAXR + ((wid >> 2) << 4);
  const int dx0  = -MAXR + ((wid & 3) << 4);

  // Per-instance parameters (uniform across the block; cached loads)
  const float x0 = bboxes[4 * i + 0], y0 = bboxes[4 * i + 1];
  const float x1 = bboxes[4 * i + 2], y1 = bboxes[4 * i + 3];
  const float cx = (x0 + x1) * 0.5f;
  const float cy = (y0 + y1) * 0.5f;
  const float area = (x1 - x0) * (y1 - y0);
  const float GSCALE = (10.0f - 1.0f) / (100000.0f - 20.0f);
  float sigma = 1.0f + GSCALE * (area - 20.0f);
  sigma = fminf(fmaxf(sigma, 1.0f), 10.0f);
  const float radius = ceilf(3.0f * sigma);
  const float neg_inv2s2 = -1.0f / (2.0f * sigma * sigma);
  const int cls   = classes[i];
  const int img   = img_id[i];
  const int noise = (int)noise_mask[i];

  // Wave-uniform tile rejection: tile box vs radius box. Force a scalar
  // branch (readfirstlane -> SGPR) so EXEC is never partially masked around
  // the WMMA (WMMA requires EXEC all-1s).
  int skip = ((float)dy0 > radius) | ((float)(dy0 + 15) < -radius) |
             ((float)dx0 > radius) | ((float)(dx0 + 15) < -radius);
  if (__builtin_amdgcn_readfirstlane(skip)) return;

  // Rank-1 outer product via WMMA: D(16x16) = gy(16x1) * gx(1x16).
  // A 16x4 f32 layout: VGPR0 lanes0-15 holds K=0 -> gaussian column there,
  // everything else zero. B 4x16: K=0 row lives in VGPR0 lanes0-15.
  const int t16 = lane & 15;
  const float gA = gauss1d(dy0 + t16, radius, neg_inv2s2);
  const float gB = gauss1d(dx0 + t16, radius, neg_inv2s2);
  const bool lo = lane < 16;
  v2f a; a.x = lo ? gA : 0.0f; a.y = 0.0f;
  v2f b; b.x = lo ? gB : 0.0f; b.y = 0.0f;
  v8f cacc = {0.f, 0.f, 0.f, 0.f, 0.f, 0.f, 0.f, 0.f};
  v8f d = __builtin_amdgcn_wmma_f32_16x16x4_f32(
      /*neg_a=*/false, a, /*neg_b=*/false, b,
      /*c_mod=*/(short)0, cacc, /*reuse_a=*/false, /*reuse_b=*/false);

  // C/D layout: VGPR v, lanes0-15 -> M=v, lanes16-31 -> M=v+8; N = lane&15.
  const int mbase = (lane >> 4) << 3;
#pragma unroll
  for (int v = 0; v < 8; ++v) {
    float val = d[v];
    const int dy = dy0 + mbase + v;
    const int dx = dx0 + t16;
    if (noise && dy == 0 && dx == 0) val = 0.9999f;  // noise: center tap only
    if (val > 0.0f) {                                // in-radius taps only
      // truncate-toward-zero, matching jnp astype(int32) on cx+dx
      const int tx = (int)(cx + (float)dx);
      const int ty = (int)(cy + (float)dy);
      if (((unsigned)tx < (unsigned)HM_W) && ((unsigned)ty < (unsigned)HM_H)) {
        const int idx = ((img * HM_C + cls) * HM_H + ty) * HM_W + tx;
        // scatter-max: non-negative IEEE f32 order == u32 bit-pattern order
        atomicMax((unsigned int*)out + idx, __float_as_uint(val));
      }
    }
  }
}

extern "C" void kernel_launch(void* const* d_in, const int* in_sizes, int n_in,
                              void* d_out, int out_size, void* d_ws, size_t ws_size,
                              hipStream_t stream) {
  // inputs: 0=main_kf_logits (unused, shape only), 1=bboxes(N,4) f32,
  //         2=classes(N) i32, 3=img_id(N) i32, 4=noise_mask(N) bool(u8)
  const float*         bboxes  = (const float*)d_in[1];
  const int*           classes = (const int*)d_in[2];
  const int*           img_id  = (const int*)d_in[3];
  const unsigned char* noise   = (const unsigned char*)d_in[4];
  float*               out     = (float*)d_out;

  const int n4 = out_size / 4;  // 11,796,480 / 4 -> float4 fill
  hm_zero<<<(n4 + 255) / 256, 256, 0, stream>>>((float4*)d_out, n4);

  const int N = in_sizes[2];    // 256 instances
  hm_scatter<<<N, 512, 0, stream>>>(bboxes, classes, img_id, noise, out);
}